// MyNet3_49469433315419
// MI455X (gfx1250) — compile-verified
//
#include <hip/hip_runtime.h>
#include <math.h>

typedef __attribute__((ext_vector_type(2))) float v2f;
typedef __attribute__((ext_vector_type(8))) float v8f;

#define NPG 37          // nodes per graph
#define KPAD 40         // 37 padded to multiple of 4 for 16x16x4 WMMA
#define NCHUNK 10       // KPAD / 4 WMMA accumulation steps

// ---------------------------------------------------------------------------
// Kernel 1: nodes[i] = x[i] * root + conv_bias   (fuses the x@root term)
// ---------------------------------------------------------------------------
__global__ void init_nodes_kernel(const float* __restrict__ x,
                                  const float* __restrict__ root,
                                  const float* __restrict__ cbias,
                                  float* __restrict__ nodes, int N)
{
    const float r = root[0];
    const float cb = cbias[0];
    const int n4 = N >> 2;
    int tid = blockIdx.x * blockDim.x + threadIdx.x;
    if (tid < n4) {
        float4 v = ((const float4*)x)[tid];
        float4 o;
        o.x = v.x * r + cb;
        o.y = v.y * r + cb;
        o.z = v.z * r + cb;
        o.w = v.w * r + cb;
        ((float4*)nodes)[tid] = o;
    }
    if (tid == 0) {                       // tail (N % 4)
        for (int i = n4 << 2; i < N; ++i) nodes[i] = x[i] * r + cb;
    }
}

// ---------------------------------------------------------------------------
// Kernel 2: edge MLP + gather + scatter-add.
//   ew = relu(ea*W1 + b1) @ W2 + b2 ;  agg[dst] += x[src] * ew
// x and agg (29.6 MB each) are L2-resident on MI455X (192 MB L2), so the
// gather and the f32 atomics run at L2 speed; the streamed index/attr
// arrays (~710 MB) set the ~30us HBM floor. 4 edges/thread, float4/int4.
// ---------------------------------------------------------------------------
__device__ __forceinline__ float edge_weight(float ea,
                                             const float* w1, const float* b1r,
                                             const float* w2, float b2r)
{
    float acc = b2r;
#pragma unroll
    for (int j = 0; j < 4; ++j) {
        float t = fmaf(ea, w1[j], b1r[j]);
        t = t > 0.f ? t : 0.f;
        acc = fmaf(t, w2[j], acc);
    }
    return acc;
}

__global__ void edge_kernel(const float* __restrict__ x,
                            const float* __restrict__ edge_attr,
                            const int* __restrict__ src,
                            const int* __restrict__ dst,
                            const float* __restrict__ W1,
                            const float* __restrict__ b1,
                            const float* __restrict__ W2,
                            const float* __restrict__ b2,
                            float* __restrict__ agg, int E)
{
    float w1[4], b1r[4], w2[4];
#pragma unroll
    for (int j = 0; j < 4; ++j) { w1[j] = W1[j]; b1r[j] = b1[j]; w2[j] = W2[j]; }
    const float b2r = b2[0];

    const int e4 = E >> 2;
    const int tid = blockIdx.x * blockDim.x + threadIdx.x;

    if (tid < e4) {
        float4 ea = ((const float4*)edge_attr)[tid];
        int4 s = ((const int4*)src)[tid];
        int4 d = ((const int4*)dst)[tid];

        float m0 = x[s.x] * edge_weight(ea.x, w1, b1r, w2, b2r);
        float m1 = x[s.y] * edge_weight(ea.y, w1, b1r, w2, b2r);
        float m2 = x[s.z] * edge_weight(ea.z, w1, b1r, w2, b2r);
        float m3 = x[s.w] * edge_weight(ea.w, w1, b1r, w2, b2r);

        (void)__hip_atomic_fetch_add(&agg[d.x], m0, __ATOMIC_RELAXED, __HIP_MEMORY_SCOPE_AGENT);
        (void)__hip_atomic_fetch_add(&agg[d.y], m1, __ATOMIC_RELAXED, __HIP_MEMORY_SCOPE_AGENT);
        (void)__hip_atomic_fetch_add(&agg[d.z], m2, __ATOMIC_RELAXED, __HIP_MEMORY_SCOPE_AGENT);
        (void)__hip_atomic_fetch_add(&agg[d.w], m3, __ATOMIC_RELAXED, __HIP_MEMORY_SCOPE_AGENT);
    }
    if (tid == 0) {                       // tail (E % 4)
        for (int e = e4 << 2; e < E; ++e) {
            float m = x[src[e]] * edge_weight(edge_attr[e], w1, b1r, w2, b2r);
            (void)__hip_atomic_fetch_add(&agg[dst[e]], m, __ATOMIC_RELAXED, __HIP_MEMORY_SCOPE_AGENT);
        }
    }
}

// ---------------------------------------------------------------------------
// Kernel 3: per-graph MLP + arctan head.
// Layer 1 (37->8) via V_WMMA_F32_16X16X4_F32: each wave owns 16 graphs
// (M=16), K padded 37->40 -> 10 WMMA accumulation steps, N=16 (cols 8..15
// zero-padded). Both A tile and a fragment-ordered, zero-padded Wa are
// staged in LDS so every fragment is one unconditional 8B-aligned
// ds_load_b64 (no divergent global loads inside the WMMA chain). All 20
// fragments are preloaded, then 10 WMMAs issue back-to-back behind a
// single LDS wait. Layers 2/3 (8->8->2, 80 MACs/graph) are scalar VALU.
// ISA 32-bit A/B fragment layout: lanes 0-15 hold K = k0,k0+1 in the two
// VGPRs; lanes 16-31 hold K = k0+2,k0+3. C/D: lane n = l%16, VGPR r holds
// row m = r + 8*(l/16).
// ---------------------------------------------------------------------------
__global__ __launch_bounds__(128)
void mlp_kernel(const float* __restrict__ nodes,
                const float* __restrict__ Wa, const float* __restrict__ ba,
                const float* __restrict__ Wb, const float* __restrict__ bb,
                const float* __restrict__ Wc, const float* __restrict__ bc,
                float* __restrict__ out, int ngraphs)
{
    __shared__ float tileA[4][16 * KPAD];      // 10240 B (per wave)
    __shared__ float tileB[NCHUNK * 64];       //  2560 B (shared, frag order)
    __shared__ float tileH[4][16 * 8];         //  2048 B

    const int lane = threadIdx.x & 31;
    const int wave = threadIdx.x >> 5;
    const int g0 = (blockIdx.x * 4 + wave) * 16;

    // Stage zero-padded Wa in per-lane fragment order:
    // tileB[kk*64 + l*2 + p] = Wa_pad[kk*4 + (l>>4)*2 + p][l&15]
    for (int i = threadIdx.x; i < NCHUNK * 64; i += 128) {
        const int kk  = i >> 6;
        const int rem = i & 63;
        const int l   = rem >> 1;
        const int p   = rem & 1;
        const int k   = kk * 4 + ((l >> 4) << 1) + p;
        const int n   = l & 15;
        tileB[i] = (k < NPG && n < 8) ? Wa[k * 8 + n] : 0.f;
    }

    // Stage 16 x KPAD node tile into LDS (coalesced global reads, zero pad).
    float* At = tileA[wave];
    for (int i = lane; i < 16 * KPAD; i += 32) {
        int gi = i / KPAD;
        int k  = i - gi * KPAD;
        float v = 0.f;
        int g = g0 + gi;
        if (k < NPG && g < ngraphs) v = nodes[(long)g * NPG + k];
        At[i] = v;
    }
    __syncthreads();

    const int m  = lane & 15;     // A: row index / B,C: col index
    const int hi = lane >> 4;     // lane-half selects K sub-pair

    // Preload all fragments (8B-aligned: k0 even, KPAD even, lane*2 even).
    v2f afrag[NCHUNK], bfrag[NCHUNK];
#pragma unroll
    for (int kk = 0; kk < NCHUNK; ++kk) {
        const int k0 = kk * 4 + hi * 2;
        afrag[kk] = *(const v2f*)&At[m * KPAD + k0];
        bfrag[kk] = *(const v2f*)&tileB[kk * 64 + lane * 2];
    }

    v8f c = {};
#pragma unroll
    for (int kk = 0; kk < NCHUNK; ++kk) {
        c = __builtin_amdgcn_wmma_f32_16x16x4_f32(
                /*neg_a=*/false, afrag[kk], /*neg_b=*/false, bfrag[kk],
                /*c_mod=*/(short)0, c, /*reuse_a=*/false, /*reuse_b=*/false);
    }

    // bias + relu, scatter the valid 16x8 h tile to LDS.
    if (m < 8) {
        const float bias = ba[m];
        float* Ht = tileH[wave];
#pragma unroll
        for (int r = 0; r < 8; ++r) {
            const int row = r + 8 * hi;
            float v = c[r] + bias;
            Ht[row * 8 + m] = v > 0.f ? v : 0.f;
        }
    }
    __syncthreads();

    // Layers 2 & 3 + angle head: one graph per lane (lanes 0..15).
    if (lane < 16 && (g0 + lane) < ngraphs) {
        const float* h = &tileH[wave][lane * 8];
        float h2[8];
#pragma unroll
        for (int j = 0; j < 8; ++j) {
            float acc = bb[j];
#pragma unroll
            for (int k = 0; k < 8; ++k) acc = fmaf(h[k], Wb[k * 8 + j], acc);
            h2[j] = acc > 0.f ? acc : 0.f;
        }
        float X = bc[0], Y = bc[1];
#pragma unroll
        for (int k = 0; k < 8; ++k) {
            X = fmaf(h2[k], Wc[k * 2 + 0], X);
            Y = fmaf(h2[k], Wc[k * 2 + 1], Y);
        }
        const float sgn = (Y > 0.f) ? 1.f : ((Y < 0.f) ? -1.f : 0.f);
        const float ang = atanf(X / Y) + 1.57079632679489662f * sgn
                        + 3.14159265358979324f;
        out[g0 + lane] = ang * (12.0f / 6.28318530717958648f);
    }
}

// ---------------------------------------------------------------------------
extern "C" void kernel_launch(void* const* d_in, const int* in_sizes, int n_in,
                              void* d_out, int out_size, void* d_ws, size_t ws_size,
                              hipStream_t stream)
{
    const float* x         = (const float*)d_in[0];
    const float* edge_attr = (const float*)d_in[1];
    const float* W1        = (const float*)d_in[2];
    const float* b1        = (const float*)d_in[3];
    const float* W2        = (const float*)d_in[4];
    const float* b2        = (const float*)d_in[5];
    const float* root      = (const float*)d_in[6];
    const float* cbias     = (const float*)d_in[7];
    const float* Wa        = (const float*)d_in[8];
    const float* ba        = (const float*)d_in[9];
    const float* Wb        = (const float*)d_in[10];
    const float* bb        = (const float*)d_in[11];
    const float* Wc        = (const float*)d_in[12];
    const float* bc        = (const float*)d_in[13];
    const int*   eidx      = (const int*)d_in[14];

    const int N = in_sizes[0];          // 7,400,000
    const int E = in_sizes[1];          // 59,200,000
    const int batch = N / NPG;          // 200,000

    float* nodes = (float*)d_ws;        // N floats = 29.6 MB scratch

    {   // nodes = x*root + conv_bias (also initializes the accumulator)
        int work = (N >> 2);
        int blocks = (work + 255) / 256;
        if (blocks < 1) blocks = 1;
        init_nodes_kernel<<<blocks, 256, 0, stream>>>(x, root, cbias, nodes, N);
    }
    {   // scatter-add edge messages into nodes
        int work = (E >> 2);
        int blocks = (work + 255) / 256;
        if (blocks < 1) blocks = 1;
        edge_kernel<<<blocks, 256, 0, stream>>>(x, edge_attr,
                                                eidx, eidx + E,
                                                W1, b1, W2, b2, nodes, E);
    }
    {   // per-graph MLP (WMMA) + angle head
        int blocks = (batch + 63) / 64;
        mlp_kernel<<<blocks, 128, 0, stream>>>(nodes, Wa, ba, Wb, bb, Wc, bc,
                                               (float*)d_out, batch);
    }
}